// ChebyNet_13872744366285
// MI455X (gfx1250) — compile-verified
//
#include <hip/hip_runtime.h>

// ChebyNet forward for MI455X (gfx1250).
// Bandwidth-bound (sparse lhat scatter ~26 GB vs 33 GFLOP dense), so we keep
// full f32 and use V_WMMA_F32_16X16X4_F32 for all feature transforms.

#define HID 64
#define OUT_DIM 10
#define NUM_GRAPHS 64
#define BN_EPS 1e-5f
#define KCHEB 5
#define NUM_STEP_PROP 6

typedef __attribute__((ext_vector_type(2))) float v2f;
typedef __attribute__((ext_vector_type(8))) float v8f;

// ---------------- degree / normalization ----------------
__global__ void deg_kernel(const int* __restrict__ row, float* __restrict__ deg, int E) {
    int e = blockIdx.x * blockDim.x + threadIdx.x;
    if (e < E) atomicAdd(&deg[row[e]], 1.0f);
}

__global__ void dinv_kernel(const float* __restrict__ deg, float* __restrict__ dinv, int N) {
    int i = blockIdx.x * blockDim.x + threadIdx.x;
    if (i < N) {
        float d = deg[i];
        dinv[i] = (d > 0.0f) ? rsqrtf(fmaxf(d, 1.0f)) : 0.0f;
    }
}

__global__ void norm_kernel(const int* __restrict__ row, const int* __restrict__ col,
                            const float* __restrict__ dinv, float* __restrict__ nrm, int E) {
    int e = blockIdx.x * blockDim.x + threadIdx.x;
    if (e < E) nrm[e] = -dinv[row[e]] * dinv[col[e]];
}

// ---------------- lhat: out[col[e],:] += norm[e] * v[row[e],:] ----------------
// grid.y indexes float4 feature quads; scatter targets stay L2-resident (<=51MB).
__global__ void lhat_kernel(const int* __restrict__ row, const int* __restrict__ col,
                            const float* __restrict__ nrm, const float* __restrict__ v,
                            float* __restrict__ out, int E, int D) {
    int e = blockIdx.x * blockDim.x + threadIdx.x;
    if (e >= E) return;
    int fq = blockIdx.y << 2;
    float nm = nrm[e];
    const float4 s = *reinterpret_cast<const float4*>(v + (size_t)row[e] * D + fq);
    float* dst = out + (size_t)col[e] * D + fq;
    atomicAdd(dst + 0, nm * s.x);
    atomicAdd(dst + 1, nm * s.y);
    atomicAdd(dst + 2, nm * s.z);
    atomicAdd(dst + 3, nm * s.w);
}

// ---------------- Chebyshev recurrence: t = 2*t - prev ----------------
__global__ void cheb_comb_kernel(float* __restrict__ t, const float* __restrict__ prev, int n) {
    int i = blockIdx.x * blockDim.x + threadIdx.x;
    if (i < n) t[i] = 2.0f * t[i] - prev[i];
}

// ---------------- accumulator init with bias ----------------
__global__ void init_bias_kernel(float* __restrict__ acc, const float* __restrict__ b, int n) {
    int i = blockIdx.x * blockDim.x + threadIdx.x;
    if (i < n) acc[i] = b[i & (HID - 1)];
}

// ---------------- WMMA f32 GEMM-accumulate: acc[16M x 16N] += A[16M x D] @ W[D x 64] ----
// One wave per 16x16 tile; K stepped by 4 via V_WMMA_F32_16X16X4_F32.
// ISA 7.12.2 32-bit layouts: lane L -> m=n=L&15, kh=L>>4;
//   A vgpr pair = {A[m][k+2kh], A[m][k+2kh+1]}; B pair = {B[k+2kh][n], B[k+2kh+1][n]};
//   C/D vgpr r <-> row (r + 8*kh).
__global__ void gemm_acc_wmma(const float* __restrict__ A, const float* __restrict__ W,
                              float* __restrict__ acc, int n_tiles, int D) {
    int wv   = threadIdx.x >> 5;
    int lane = threadIdx.x & 31;
    int tm = blockIdx.x * 8 + wv;
    if (tm >= n_tiles) return;              // whole-wave uniform exit: EXEC stays all-1s
    int m  = lane & 15;
    int kh = lane >> 4;
    int col = (blockIdx.y << 4) + m;        // n == m == lane&15
    size_t row0 = (size_t)tm * 16;
    const float* Arow = A + (row0 + m) * (size_t)D;

    v8f c;
#pragma unroll
    for (int r = 0; r < 8; ++r)
        c[r] = acc[(row0 + r + 8 * kh) * HID + col];

#pragma unroll 4
    for (int kb = 0; kb < D; kb += 4) {
        int k0 = kb + 2 * kh;
        v2f a, b;
        a.x = Arow[k0];
        a.y = Arow[k0 + 1];
        b.x = W[(size_t)k0 * HID + col];
        b.y = W[(size_t)(k0 + 1) * HID + col];
        c = __builtin_amdgcn_wmma_f32_16x16x4_f32(false, a, false, b, (short)0, c, false, false);
    }

#pragma unroll
    for (int r = 0; r < 8; ++r)
        acc[(row0 + r + 8 * kh) * HID + col] = c[r];
}

// ---------------- fused ReLU + eval-mode BatchNorm ----------------
__global__ void relu_bn_kernel(const float* __restrict__ acc,
                               const float* __restrict__ gamma, const float* __restrict__ beta,
                               const float* __restrict__ mean,  const float* __restrict__ var,
                               float* __restrict__ h, int n) {
    int i = blockIdx.x * blockDim.x + threadIdx.x;
    if (i >= n) return;
    int j = i & (HID - 1);
    float v = fmaxf(acc[i], 0.0f);
    h[i] = (v - mean[j]) * rsqrtf(var[j] + BN_EPS) * gamma[j] + beta[j];
}

// ---------------- global mean pool (atomic accumulate) ----------------
__global__ void pool_kernel(const float* __restrict__ h, const int* __restrict__ batch,
                            float* __restrict__ psum, float* __restrict__ pcnt, int n) {
    int i = blockIdx.x * blockDim.x + threadIdx.x;
    if (i >= n) return;
    int node = i >> 6;
    int j = i & (HID - 1);
    int g = batch[node];
    atomicAdd(&psum[g * HID + j], h[i]);
    if (j == 0) atomicAdd(&pcnt[g], 1.0f);
}

// ---------------- final tiny linear: out = (psum/cnt) @ lin2_W + lin2_b ----------------
__global__ void final_kernel(const float* __restrict__ psum, const float* __restrict__ pcnt,
                             const float* __restrict__ lw, const float* __restrict__ lb,
                             float* __restrict__ out) {
    int i = blockIdx.x * blockDim.x + threadIdx.x;
    if (i >= NUM_GRAPHS * OUT_DIM) return;
    int g = i / OUT_DIM, o = i % OUT_DIM;
    float inv = 1.0f / fmaxf(pcnt[g], 1.0f);
    float a = lb[o];
    for (int j = 0; j < HID; ++j)
        a += psum[g * HID + j] * inv * lw[j * OUT_DIM + o];
    out[i] = a;
}

// ---------------- one ChebConv (K=5): acc = sum_k T_k @ W[k] + b ----------------
static void run_cheb(const float* hin, int D, const float* W, const float* b,
                     float* acc, float* t1, float* t2, float* t3,
                     const int* row, const int* col, const float* nrm,
                     int N, int E, hipStream_t stream) {
    int tilesM = N / 16;                         // N = 100000 = 6250*16 exactly
    dim3 gGemm((tilesM + 7) / 8, HID / 16);
    dim3 gLhat((E + 255) / 256, D / 4);
    size_t bytesT = (size_t)N * D * sizeof(float);
    int nh = N * HID, nd = N * D;

    init_bias_kernel<<<(nh + 255) / 256, 256, 0, stream>>>(acc, b, nh);
    // k = 0: T0 = h
    gemm_acc_wmma<<<gGemm, 256, 0, stream>>>(hin, W, acc, tilesM, D);
    // k = 1: T1 = lhat(h)
    hipMemsetAsync(t1, 0, bytesT, stream);
    lhat_kernel<<<gLhat, 256, 0, stream>>>(row, col, nrm, hin, t1, E, D);
    gemm_acc_wmma<<<gGemm, 256, 0, stream>>>(t1, W + (size_t)1 * D * HID, acc, tilesM, D);

    const float* P0 = hin;
    const float* P1 = t1;
    float* freeb[3] = {t2, t3, t1};              // rotation keeps P0/P1 live
    for (int k = 2; k < KCHEB; ++k) {
        float* tn = freeb[k - 2];
        hipMemsetAsync(tn, 0, bytesT, stream);
        lhat_kernel<<<gLhat, 256, 0, stream>>>(row, col, nrm, P1, tn, E, D);
        cheb_comb_kernel<<<(nd + 255) / 256, 256, 0, stream>>>(tn, P0, nd);
        gemm_acc_wmma<<<gGemm, 256, 0, stream>>>(tn, W + (size_t)k * D * HID, acc, tilesM, D);
        P0 = P1; P1 = tn;
    }
}

extern "C" void kernel_launch(void* const* d_in, const int* in_sizes, int n_in,
                              void* d_out, int out_size, void* d_ws, size_t ws_size,
                              hipStream_t stream) {
    const float* x     = (const float*)d_in[0];
    const int*   eidx  = (const int*)d_in[1];
    const int*   batch = (const int*)d_in[2];
    const float* W0    = (const float*)d_in[3];
    const float* b0    = (const float*)d_in[4];
    const float* W1    = (const float*)d_in[5];
    const float* b1    = (const float*)d_in[6];
    const float* gamma = (const float*)d_in[7];
    const float* beta  = (const float*)d_in[8];
    const float* mean  = (const float*)d_in[9];
    const float* var   = (const float*)d_in[10];
    const float* lw    = (const float*)d_in[11];
    const float* lb    = (const float*)d_in[12];
    float* out = (float*)d_out;

    const int IN = 128;
    int N = in_sizes[0] / IN;
    int E = in_sizes[1] / 2;
    const int* row = eidx;
    const int* col = eidx + E;

    // carve workspace (256B aligned slices)
    char* p = (char*)d_ws;
    auto carve = [&](size_t nbytes) {
        float* r = (float*)p;
        p += (nbytes + 255) & ~(size_t)255;
        return r;
    };
    float* deg  = carve((size_t)N * 4);
    float* dinv = carve((size_t)N * 4);
    float* nrm  = carve((size_t)E * 4);
    float* t1   = carve((size_t)N * IN * 4);
    float* t2   = carve((size_t)N * IN * 4);
    float* t3   = carve((size_t)N * IN * 4);
    float* h    = carve((size_t)N * HID * 4);
    float* acc  = carve((size_t)N * HID * 4);
    float* psum = carve((size_t)NUM_GRAPHS * HID * 4);
    float* pcnt = carve((size_t)NUM_GRAPHS * 4);

    int nh = N * HID;

    // edge normalization: norm = -dinv[row]*dinv[col]
    hipMemsetAsync(deg, 0, (size_t)N * 4, stream);
    deg_kernel<<<(E + 255) / 256, 256, 0, stream>>>(row, deg, E);
    dinv_kernel<<<(N + 255) / 256, 256, 0, stream>>>(deg, dinv, N);
    norm_kernel<<<(E + 255) / 256, 256, 0, stream>>>(row, col, dinv, nrm, E);

    // layer 0: 128 -> 64
    run_cheb(x, IN, W0, b0, acc, t1, t2, t3, row, col, nrm, N, E, stream);
    relu_bn_kernel<<<(nh + 255) / 256, 256, 0, stream>>>(acc, gamma, beta, mean, var, h, nh);

    // 6 propagation layers: 64 -> 64
    for (int s = 0; s < NUM_STEP_PROP; ++s) {
        run_cheb(h, HID, W1, b1, acc, t1, t2, t3, row, col, nrm, N, E, stream);
        relu_bn_kernel<<<(nh + 255) / 256, 256, 0, stream>>>(acc, gamma, beta, mean, var, h, nh);
    }

    // global mean pool + final linear
    hipMemsetAsync(psum, 0, (size_t)NUM_GRAPHS * HID * 4, stream);
    hipMemsetAsync(pcnt, 0, (size_t)NUM_GRAPHS * 4, stream);
    pool_kernel<<<(nh + 255) / 256, 256, 0, stream>>>(h, batch, psum, pcnt, nh);
    final_kernel<<<(NUM_GRAPHS * OUT_DIM + 255) / 256, 256, 0, stream>>>(psum, pcnt, lw, lb, out);
}